// SO2Tensorproduct_84748294684804
// MI455X (gfx1250) — compile-verified
//
#include <hip/hip_runtime.h>

typedef __attribute__((ext_vector_type(2))) float v2f;
typedef __attribute__((ext_vector_type(4))) float v4f;
typedef __attribute__((ext_vector_type(8))) float v8f;

// PERM from the reference (L=4): permuted index q <-> raw coefficient PERM[q]
// (same mapping for input gather and output scatter).
__constant__ int PERM[25] = {
    0, 2, 6, 12, 20,              // m=0
    3, 7, 13, 21, 1, 5, 11, 19,   // m=1
    8, 14, 22, 4, 10, 18,         // m=2
    15, 23, 9, 17,                // m=3
    24, 16                        // m=4
};

#define ROWLEN 800       // 25 * 32 floats per edge
#define EDGES_PER_WG 128

// One m-block phase. K == N (C == MO == 32). HALF = size*C; m=0 has HALF==K.
// Weights staged TRANSPOSED in LDS: wsT[n*S + k], S = HALF+4 (bank-conflict-free
// for the half-wave-split b64 fragment reads). Complex structure handled by
// A-side sign (a1n = -blk1) and column selection, so B reads are raw w columns.
template <int K, int HALF, int QOFF, bool BIAS>
__device__ __forceinline__ void run_phase(const float* __restrict__ xg,
                                          const float* __restrict__ wg,
                                          const float* __restrict__ bg,
                                          float* __restrict__ outg,
                                          float* smem, int ebase, int E) {
  constexpr int S = HALF + 4;      // wsT row stride (per output column n)
  constexpr int XSTRIDE = K + 4;
  constexpr int H4 = HALF / 4;     // fragments per k-half
  constexpr int T_HALF = HALF / 16;
  constexpr int NT = K / 16;

  float* wsT = smem;               // K * S floats
  float* xs = smem + K * S;        // 128 * XSTRIDE floats

  const int tid = threadIdx.x;

  // ---- stage weights transposed (128-bit global reads, scalar LDS scatter) ----
  for (int i = tid * 4; i < HALF * K; i += 256 * 4) {
    const int r = i / K;           // k row in w (K multiple of 4 -> same row)
    const int c = i - r * K;       // n col in w, multiple of 4
    const v4f wv = *(const v4f*)(wg + i);
    wsT[(c + 0) * S + r] = wv.x;
    wsT[(c + 1) * S + r] = wv.y;
    wsT[(c + 2) * S + r] = wv.z;
    wsT[(c + 3) * S + r] = wv.w;
  }
  // ---- stage permuted input slab: 128 edges x K, 128-bit moves (zero tail) ----
  for (int i = tid * 4; i < EDGES_PER_WG * K; i += 256 * 4) {
    const int el = i / K;
    const int rc = i - el * K;     // qr*32 + c, multiple of 4 (within one coef row)
    const int qr = rc >> 5;
    const int e = ebase + el;
    v4f v = {0.f, 0.f, 0.f, 0.f};
    if (e < E)
      v = *(const v4f*)(xg + (size_t)e * ROWLEN + (size_t)PERM[QOFF + qr] * 32 +
                        (rc & 31));
    *(v4f*)(xs + el * XSTRIDE + rc) = v;
  }
  __syncthreads();

  const int lane = tid & 31;
  const int wave = tid >> 5;
  const int lane16 = lane & 15;
  const int h = lane >> 4;         // half-wave id
  const int eloc = wave * 16;

  // ---- A fragments in registers ----
  // A frag layout: VGPR0 lane l -> A[l%16][kb + 2*(l/16)], VGPR1 -> +1
  v2f a0[H4];                                   // blk0 (or whole block for m=0)
  v2f a1[(HALF < K) ? H4 : 1];                  // blk1
  v2f a1n[(HALF < K) ? H4 : 1];                 // -blk1
  const v2f* ap = (const v2f*)(xs + (eloc + lane16) * XSTRIDE + 2 * h);
#pragma unroll
  for (int i = 0; i < H4; ++i) a0[i] = ap[2 * i];
  if constexpr (HALF < K) {
#pragma unroll
    for (int i = 0; i < H4; ++i) {
      a1[i] = ap[HALF / 2 + 2 * i];
      a1n[i].x = -a1[i].x;
      a1n[i].y = -a1[i].y;
    }
  }

  auto store_tile = [&](int t, v8f acc) {
    // C layout: VGPR p holds rows p (lanes 0-15) / p+8 (lanes 16-31), N = lane%16
    const int q = QOFF + (t >> 1);
    const int outcol = PERM[q] * 32 + ((t & 1) << 4) + lane16;
#pragma unroll
    for (int p = 0; p < 8; ++p) {
      const int e = ebase + eloc + p + 8 * h;
      if (e < E) outg[(size_t)e * ROWLEN + outcol] = acc[p];
    }
  };

  // ---- group 0: output cols n < HALF  (out_r = blk0*Wr - blk1*Wi) ----
#pragma unroll 1
  for (int t = 0; t < T_HALF; ++t) {
    const int ncol = t * 16 + lane16;
    const v2f* bp0 = (const v2f*)(wsT + ncol * S + 2 * h);    // Wr column n
    v8f acc = {0.f, 0.f, 0.f, 0.f, 0.f, 0.f, 0.f, 0.f};
#pragma unroll
    for (int i = 0; i < H4; ++i) {
      const v2f b = bp0[2 * i];                                // b64, 2/inst merge
      acc = __builtin_amdgcn_wmma_f32_16x16x4_f32(
          false, a0[i], false, b, (short)0, acc, false, false);
    }
    if constexpr (HALF < K) {
      const v2f* bp1 = (const v2f*)(wsT + (ncol + HALF) * S + 2 * h); // Wi col n
#pragma unroll
      for (int i = 0; i < H4; ++i) {
        const v2f b = bp1[2 * i];
        acc = __builtin_amdgcn_wmma_f32_16x16x4_f32(
            false, a1n[i], false, b, (short)0, acc, false, false);
      }
    }
    if constexpr (BIAS) {
      const float bv = bg[ncol];
#pragma unroll
      for (int p = 0; p < 8; ++p) acc[p] += bv;
    }
    store_tile(t, acc);
  }

  // ---- group 1: output cols n >= HALF  (out_i = blk1*Wr + blk0*Wi) ----
  if constexpr (HALF < K) {
#pragma unroll 1
    for (int t = T_HALF; t < NT; ++t) {
      const int ncol = t * 16 + lane16;
      const v2f* bp0 = (const v2f*)(wsT + (ncol - HALF) * S + 2 * h); // Wr n-HALF
      const v2f* bp1 = (const v2f*)(wsT + ncol * S + 2 * h);          // Wi n-HALF
      v8f acc = {0.f, 0.f, 0.f, 0.f, 0.f, 0.f, 0.f, 0.f};
#pragma unroll
      for (int i = 0; i < H4; ++i) {
        const v2f b = bp0[2 * i];
        acc = __builtin_amdgcn_wmma_f32_16x16x4_f32(
            false, a1[i], false, b, (short)0, acc, false, false);
      }
#pragma unroll
      for (int i = 0; i < H4; ++i) {
        const v2f b = bp1[2 * i];
        acc = __builtin_amdgcn_wmma_f32_16x16x4_f32(
            false, a0[i], false, b, (short)0, acc, false, false);
      }
      store_tile(t, acc);
    }
  }
}

__global__ void __launch_bounds__(256, 1)
so2_tensor_product_kernel(const float* __restrict__ x,
                          const float* __restrict__ w0,
                          const float* __restrict__ b0,
                          const float* __restrict__ w1,
                          const float* __restrict__ w2,
                          const float* __restrict__ w3,
                          const float* __restrict__ w4,
                          float* __restrict__ out, int E) {
  extern __shared__ float smem[];
  const int ebase = blockIdx.x * EDGES_PER_WG;

  run_phase<160, 160, 0, true>(x, w0, b0, out, smem, ebase, E);        // m=0
  __syncthreads();
  run_phase<256, 128, 5, false>(x, w1, nullptr, out, smem, ebase, E);  // m=1
  __syncthreads();
  run_phase<192, 96, 13, false>(x, w2, nullptr, out, smem, ebase, E);  // m=2
  __syncthreads();
  run_phase<128, 64, 19, false>(x, w3, nullptr, out, smem, ebase, E);  // m=3
  __syncthreads();
  run_phase<64, 32, 23, false>(x, w4, nullptr, out, smem, ebase, E);   // m=4
}

extern "C" void kernel_launch(void* const* d_in, const int* in_sizes, int n_in,
                              void* d_out, int out_size, void* d_ws, size_t ws_size,
                              hipStream_t stream) {
  const float* x  = (const float*)d_in[0];
  const float* w0 = (const float*)d_in[1];
  const float* b0 = (const float*)d_in[2];
  const float* w1 = (const float*)d_in[3];
  const float* w2 = (const float*)d_in[4];
  const float* w3 = (const float*)d_in[5];
  const float* w4 = (const float*)d_in[6];
  float* out = (float*)d_out;

  const int E = in_sizes[0] / ROWLEN;
  const int grid = (E + EDGES_PER_WG - 1) / EDGES_PER_WG;

  // Max LDS over phases (m=1): wsT 256*(128+4) + xs 128*(256+4) = 67072 floats.
  const int shmem_bytes = (256 * (128 + 4) + EDGES_PER_WG * (256 + 4)) * (int)sizeof(float);
  (void)hipFuncSetAttribute((const void*)so2_tensor_product_kernel,
                            hipFuncAttributeMaxDynamicSharedMemorySize, shmem_bytes);

  so2_tensor_product_kernel<<<grid, 256, shmem_bytes, stream>>>(
      x, w0, b0, w1, w2, w3, w4, out, E);
}